// APM_p_Graph_45938970198649
// MI455X (gfx1250) — compile-verified
//
#include <hip/hip_runtime.h>

typedef __attribute__((ext_vector_type(2))) float v2f;
typedef __attribute__((ext_vector_type(8))) float v8f;
typedef unsigned long long u64;

#define WAVES_PER_BLOCK 8
#define KSEL 12

// ---------------------------------------------------------------------------
// Kernel 1: pack p[n][3] into float4 (x, y, z, |p|^2)
// ---------------------------------------------------------------------------
__global__ void APM_knn_prep_kernel(const float* __restrict__ p,
                                    float4* __restrict__ pk, int n) {
    int i = blockIdx.x * blockDim.x + threadIdx.x;
    if (i < n) {
        float x = p[3 * i + 0];
        float y = p[3 * i + 1];
        float z = p[3 * i + 2];
        pk[i] = make_float4(x, y, z, x * x + y * y + z * z);
    }
}

// ---------------------------------------------------------------------------
// Kernel 2: one wave per 16-query tile, software-pipelined over candidate
// tiles. Gram tile via V_WMMA_F32_16X16X4_F32 (K=3 padded to 4); d2 mapped
// to order-preserving u32; LDS transpose (double-buffered); branchless
// sorted top-12 on packed u64 (d2,idx) keys with a cheap u32 pre-guard.
// ---------------------------------------------------------------------------
__global__ __launch_bounds__(256) void APM_knn_gcn_kernel(
        const float4* __restrict__ pk,
        const float* __restrict__ Wl,
        const float* __restrict__ bl,
        float* __restrict__ out,
        int n) {
    __shared__ __align__(16) unsigned tile[WAVES_PER_BLOCK][2][256];  // 16 KB
    __shared__ u64 listK[WAVES_PER_BLOCK][16][2][KSEL];               // 24 KB

    const int wave  = threadIdx.x >> 5;
    const int lane  = threadIdx.x & 31;
    const int l15   = lane & 15;
    const int hi    = (lane >> 4) & 1;
    const int qbase = (blockIdx.x * WAVES_PER_BLOCK + wave) * 16;

    // ---- A fragment (16x4 f32): lanes 0-15 carry K=0,1 (x,y) of row l15;
    //      lanes 16-31 carry K=2,3 (z, 0-pad). Loop invariant.
    const float4 q4 = pk[qbase + l15];
    v2f a;
    a.x = hi ? q4.z : q4.x;
    a.y = hi ? 0.0f : q4.y;

    // ---- |p|^2 of the 8 C-rows this lane sees (C VGPR r: M = r + 8*hi)
    const float* pf = (const float*)pk;
    float sqQm[8];
#pragma unroll
    for (int r = 0; r < 8; ++r) sqQm[r] = pf[4 * (qbase + r + 8 * hi) + 3];

    // Gram tile for candidate tile at jbase -> 8 order-preserving skeys
    auto gram_skeys = [&](int jbase, unsigned* skout) {
        const float4 c4 = pk[jbase + l15];
        v2f bb;
        bb.x = hi ? c4.z : c4.x;
        bb.y = hi ? 0.0f : c4.y;
        v8f c = {};
        c = __builtin_amdgcn_wmma_f32_16x16x4_f32(
                false, a, false, bb, (short)0, c, false, false);
        const float sc = c4.w;
#pragma unroll
        for (int r = 0; r < 8; ++r) {
            const float d2 = fmaf(-2.0f, c[r], sqQm[r] + sc);
            const unsigned ub = __float_as_uint(d2);
            skout[r] = ub ^ (unsigned)(((int)ub >> 31) | 0x80000000);
        }
    };

    // ---- selection state: lane owns row (lane>>1), half (lane&1)
    const int myrow = lane >> 1;
    const int half  = lane & 1;
    u64 lst[KSEL];
#pragma unroll
    for (int k = 0; k < KSEL; ++k) lst[k] = 0xFFFFFFFFFFFFFFFFull;
    unsigned h11 = 0xFFFFFFFFu;   // high 32 key bits (skey) of lst[11]

    const int ntiles = n >> 4;
    unsigned skc[8], skn[8];
    gram_skeys(0, skc);           // pipeline prologue

    int buf = 0;
    for (int jt = 0; jt < ntiles; ++jt) {
        // store current tile's skeys (transpose row ownership)
        unsigned* tb = &tile[wave][buf][0];
#pragma unroll
        for (int r = 0; r < 8; ++r) tb[(r + 8 * hi) * 16 + l15] = skc[r];

        // prefetch next tile: global b128 load + WMMA + key map, all before
        // the fence so XDL latency overlaps the insertion phase below
        const int jn = (jt + 1 < ntiles) ? (jt + 1) : (ntiles - 1);
        gram_skeys(jn << 4, skn);

        asm volatile("s_wait_dscnt 0" ::: "memory");

        // read my 8 candidates as two b128 LDS loads
        const unsigned* rowp = &tile[wave][buf][myrow * 16 + half * 8];
        const uint4 qa = *(const uint4*)(rowp);
        const uint4 qb = *(const uint4*)(rowp + 4);
        const unsigned sv[8] = {qa.x, qa.y, qa.z, qa.w, qb.x, qb.y, qb.z, qb.w};
        const unsigned cbase = (unsigned)((jt << 4) + half * 8);

#pragma unroll
        for (int t = 0; t < 8; ++t) {
            // conservative u32 pre-guard; the min/max chain is idempotent for
            // non-qualifying keys, so <= (tie on skey) is safe.
            if (sv[t] <= h11) {
                u64 nk = ((u64)sv[t] << 16) | (u64)(cbase + t);
#pragma unroll
                for (int k = 0; k < KSEL; ++k) {   // branchless sorted insert
                    const u64 cur = lst[k];
                    const bool lt = nk < cur;
                    lst[k] = lt ? nk : cur;
                    nk     = lt ? cur : nk;
                }
                h11 = (unsigned)(lst[KSEL - 1] >> 16);
            }
        }

#pragma unroll
        for (int r = 0; r < 8; ++r) skc[r] = skn[r];
        buf ^= 1;
    }

    // ---- publish per-lane sorted lists
#pragma unroll
    for (int k = 0; k < KSEL; ++k) listK[wave][myrow][half][k] = lst[k];
    asm volatile("s_wait_dscnt 0" ::: "memory");

    // ---- one lane per row: merge two sorted 12-lists (keys unique), drop
    //      entry 0 (self), evaluate closed-form GCNConv + mean pool.
    if (lane < 16) {
        const int m = lane;
        const float w0 = Wl[0], w1 = Wl[1], w2 = Wl[2];
        const float bias = bl[0];
        const float4 q = pk[qbase + m];

        int ia = 0, ib = 0;
        float acc = 0.0f;
#pragma unroll
        for (int k = 0; k < KSEL; ++k) {
            const u64 ka = listK[wave][m][0][ia];
            const u64 kb = listK[wave][m][1][ib];
            const bool tA = ka < kb;
            const u64 kk = tA ? ka : kb;
            ia += tA ? 1 : 0;
            ib += tA ? 0 : 1;
            if (k >= 1) {
                const unsigned j = (unsigned)(kk & 0xFFFFu);
                const float4 nb = pk[j];
                acc += fabsf(q.x - nb.x) * w0 +
                       fabsf(q.y - nb.y) * w1 +
                       fabsf(q.z - nb.z) * w2;
            }
        }
        const float inv_s2 = 0.70710678118654752f;
        const float c0f = (1.0f + 11.0f * inv_s2) / 12.0f;
        const float c1f = 0.5f / 12.0f;
        const float xw0 = q.x * w0 + q.y * w1 + q.z * w2;
        out[qbase + m] = c0f * xw0 + c1f * acc + bias;
    }
}

// ---------------------------------------------------------------------------
extern "C" void kernel_launch(void* const* d_in, const int* in_sizes, int n_in,
                              void* d_out, int out_size, void* d_ws, size_t ws_size,
                              hipStream_t stream) {
    const float* p  = (const float*)d_in[0];   // [2,8192,3] -> n x 3
    const float* Wl = (const float*)d_in[1];   // [1,3]
    const float* bl = (const float*)d_in[2];   // [1]
    float* out = (float*)d_out;

    const int n = in_sizes[0] / 3;             // 16384

    float4* pack = (float4*)d_ws;              // n * 16 B = 256 KB

    {
        const int threads = 256;
        const int blocks = (n + threads - 1) / threads;
        APM_knn_prep_kernel<<<blocks, threads, 0, stream>>>(p, pack, n);
    }
    {
        const int threads = 32 * WAVES_PER_BLOCK;       // 256
        const int blocks = n / (16 * WAVES_PER_BLOCK);  // 128
        APM_knn_gcn_kernel<<<blocks, threads, 0, stream>>>(pack, Wl, bl, out, n);
    }
}